// Cfe_41059887350014
// MI455X (gfx1250) — compile-verified
//
#include <hip/hip_runtime.h>
#include <stdint.h>

#ifndef __has_builtin
#define __has_builtin(x) 0
#endif

#define T_STEPS   730
#define B_TOT     2048
#define NSTATIC   210
#define LENF      15
#define NEARZERO  1e-5f
#define CT        32                      // timesteps per TDM tile
#define NCHUNK    ((T_STEPS + CT - 1) / CT)
#define BPB       16                      // basins per block
#define THREADS   256                     // 8 waves; wave = 2 basins x 16 members
#define ROW_ELEMS (B_TOT * 2)             // floats per timestep row of x_phy
#define TILE_W    (BPB * 2)               // 32 floats = 128 B per row per block
#define BUF_FLOATS (CT * TILE_W)          // 1024 floats = 4 KB per buffer

#if defined(__HIP_DEVICE_COMPILE__) && __has_builtin(__builtin_amdgcn_tensor_load_to_lds) && __has_builtin(__builtin_amdgcn_s_wait_tensorcnt)
#define USE_TDM 1
#else
#define USE_TDM 0
#endif

#if USE_TDM
typedef unsigned int u32x4 __attribute__((ext_vector_type(4)));
typedef int          i32x4 __attribute__((ext_vector_type(4)));
typedef int          i32x8 __attribute__((ext_vector_type(8)));

// Issue one TDM 2-D tile load: rows x 32 floats, row stride = ROW_ELEMS floats.
__device__ __forceinline__ void tdm_issue(const float* x, uint32_t lds_byte,
                                          int t0, int rows, int b0) {
  uint64_t ga = (uint64_t)(uintptr_t)x +
                (((uint64_t)t0 * ROW_ELEMS + (uint64_t)b0 * 2) << 2);
  u32x4 g0;
  g0[0] = 1u;                                   // count=1, user descriptor
  g0[1] = lds_byte;                             // lds_addr (bytes)
  g0[2] = (uint32_t)(ga & 0xFFFFFFFFu);         // global_addr[31:0]
  g0[3] = (uint32_t)((ga >> 32) & 0x1FFFFFFu)   // global_addr[56:32]
          | (2u << 30);                         // type=2 ("image")
  uint32_t td0 = (uint32_t)ROW_ELEMS;           // tensor_dim0 (elements)
  uint32_t td1 = (uint32_t)rows;                // tensor_dim1 (rows in tile)
  i32x8 g1;
  g1[0] = (int)(2u << 16);                      // data_size=4B, no multicast
  g1[1] = (int)((td0 & 0xFFFFu) << 16);         // abar=0 | tensor_dim0 lo16
  g1[2] = (int)((td0 >> 16) | ((td1 & 0xFFFFu) << 16));
  g1[3] = (int)((td1 >> 16) | ((uint32_t)TILE_W << 16));   // tile_dim0=32
  g1[4] = (int)(td1 & 0xFFFFu);                 // tile_dim1=rows, tile_dim2=0
  g1[5] = (int)td0;                             // tensor_dim0_stride lo32
  g1[6] = 0;                                    // stride hi16 | dim1_stride lo16
  g1[7] = 0;
  i32x4 z4 = {0, 0, 0, 0};
  i32x8 z8 = {0, 0, 0, 0, 0, 0, 0, 0};
  __builtin_amdgcn_tensor_load_to_lds(g0, g1, z4, z4, z8, 0);
}
#endif

__device__ __forceinline__ int imin(int a, int b) { return a < b ? a : b; }

__global__ void __launch_bounds__(THREADS)
cfe_scan_kernel(const float* __restrict__ x_phy,       // (T, B, 2)
                const float* __restrict__ phy_static,  // (B, 210)
                float* __restrict__ out)               // (T, B)
{
  extern __shared__ float smem[];                // 2 * BUF_FLOATS
  const int tid  = (int)threadIdx.x;
  const int wave = tid >> 5;
  const int lane = tid & 31;
  const int bw   = lane >> 4;                    // basin within wave (0/1)
  const int m    = lane & 15;                    // ensemble member
  const int b0   = (int)blockIdx.x * BPB;
  const int bloc = wave * 2 + bw;                // basin within block [0,16)
  const int b    = b0 + bloc;

  const float* ps = phy_static + (size_t)b * NSTATIC;

  // ---- parameter de-normalization (PARAM_BOUNDS order) ----
  #define PGET(i, lb, ub) (ps[(i) * 16 + m] * ((ub) - (lb)) + (lb))
  float schaake = PGET(0, 0.0f, 0.1f);
  float smax    = PGET(1, 0.3f, 0.6f);
  float sdep    = PGET(2, 0.5f, 3.0f);
  float wlt     = PGET(3, 0.05f, 0.2f);
  float satpsi  = PGET(4, 0.01f, 0.5f);
  float bb      = PGET(5, 2.0f, 12.0f);
  float mult    = PGET(6, 100.0f, 2000.0f);
  float satdk   = PGET(7, 1e-7f, 1e-4f);
  float slop    = PGET(8, 0.0f, 1.0f);
  float maxgw   = PGET(9, 0.01f, 0.5f);
  float cgw     = PGET(10, 1e-6f, 1e-3f);
  float expon   = PGET(11, 1.0f, 8.0f);
  float K       = PGET(12, 0.01f, 0.5f);
  #undef PGET

  const float maxSoil     = smax * sdep;
  float fcfrac            = powf(satpsi / sdep, 1.0f / bb);
  fcfrac                  = fminf(fmaxf(fcfrac, 0.0f), 1.0f);
  const float fcThr       = smax * fcfrac * sdep;
  const float schk        = 1.0f - expf(-schaake);   // TIMESTEP_D = 1
  const float km          = satdk * mult;
  const float klat        = km * slop;
  const float invRange    = 1.0f / (smax - wlt + NEARZERO);
  const float invSdep     = 1.0f / sdep;
  const float invMaxGw    = 1.0f / maxgw;
  const float invDenomFc  = 1.0f / fmaxf(maxSoil - fcThr, NEARZERO);

  // ---- routing weights: normalization cancels lgamma(a) and th^a ----
  const float a_g  = fmaxf(ps[208] * 2.9f, 0.0f) + 0.1f;
  const float th_g = fmaxf(ps[209] * 6.5f, 0.0f) + 0.5f;
  const float am1 = a_g - 1.0f, invTh = 1.0f / th_g;
  float w[LENF], wsum = 0.0f;
  #pragma unroll
  for (int j = 0; j < LENF; ++j) {
    float tj = (float)j + 0.5f;
    w[j] = expf(am1 * logf(tj) - tj * invTh);
    wsum += w[j];
  }
  const float invW = 1.0f / wsum;
  #pragma unroll
  for (int j = 0; j < LENF; ++j) w[j] *= invW;

  float ring[LENF];
  #pragma unroll
  for (int j = 0; j < LENF; ++j) ring[j] = 0.0f;

  float soil = 0.05f, gw = 0.01f;
  float n0 = NEARZERO, n1 = NEARZERO, n2 = NEARZERO;

#if USE_TDM
  const uint32_t lds_byte0 = (uint32_t)(uintptr_t)smem;  // flat LDS aperture: low 32 bits = LDS offset
  if (wave == 0) tdm_issue(x_phy, lds_byte0, 0, imin(CT, T_STEPS), b0);
#endif

  int t = 0;
  for (int c = 0; c < NCHUNK; ++c) {
    const int t0   = c * CT;
    const int rows = imin(CT, T_STEPS - t0);
    const int cur  = c & 1;

#if USE_TDM
    if (wave == 0) __builtin_amdgcn_s_wait_tensorcnt(0);
    __syncthreads();   // publishes buf[cur]; also fences reads of buf[1-cur]
    if (wave == 0 && (c + 1) < NCHUNK) {
      tdm_issue(x_phy, lds_byte0 + (uint32_t)((1 - cur) * BUF_FLOATS * 4),
                t0 + CT, imin(CT, T_STEPS - t0 - CT), b0);
    }
#else
    __syncthreads();
    for (int i = tid; i < rows * TILE_W; i += THREADS) {
      int r = i / TILE_W, col = i - r * TILE_W;
      smem[cur * BUF_FLOATS + i] =
          x_phy[(size_t)(t0 + r) * ROW_ELEMS + (size_t)b0 * 2 + col];
    }
    __syncthreads();
#endif
    const float* fbuf = smem + cur * BUF_FLOATS;

    for (int tl = 0; tl < rows; ++tl, ++t) {
      const float p   = fbuf[tl * TILE_W + bloc * 2 + 0];
      const float pet = fbuf[tl * TILE_W + bloc * 2 + 1];

      // ---- CFE step ----
      float et_rain = fminf(p, pet);
      float p_rem   = p - et_rain;
      float pet_rem = pet - et_rain;

      float smc    = soil * invSdep;
      float budyko = fminf(fmaxf((smc - wlt) * invRange, 0.0f), 1.0f);
      float aet_s  = fminf(pet_rem * budyko, fmaxf(soil - NEARZERO, 0.0f));
      soil -= aet_s;

      float deficit   = fmaxf(maxSoil - soil, 0.0f);
      float infil_cap = deficit * schk;
      float runoff    = p_rem * p_rem / (p_rem + infil_cap + NEARZERO);
      runoff          = fminf(runoff, p_rem);
      float infil     = fminf(p_rem - runoff, deficit);
      runoff          = p_rem - infil;
      soil += infil;

      float above = fmaxf(soil - fcThr, 0.0f);
      float frac  = above * invDenomFc;
      float perc  = km * frac;
      float lat   = klat * frac;
      float tot   = perc + lat;
      float scale = fminf(1.0f, above / fmaxf(tot, NEARZERO));
      perc *= scale;
      lat  *= scale;
      soil -= perc + lat;
      gw   += perc;

      float gfill = fminf(fmaxf(gw * invMaxGw, 0.0f), 1.0f);
      float q_gw  = cgw * expm1f(expon * gfill);
      q_gw = fminf(fmaxf(q_gw, 0.0f), fmaxf(gw - NEARZERO, 0.0f));
      gw  -= q_gw;

      float inflow = lat;
      float s  = n0 + inflow; float o = K * s; n0 = s - o; inflow = o;
      s = n1 + inflow; o = K * s; n1 = s - o; inflow = o;
      s = n2 + inflow; o = K * s; n2 = s - o; inflow = o;

      float q = runoff + inflow + q_gw;

      // ---- 16-member ensemble mean: wave32 butterfly (stays within lane[3:0]) ----
      q += __shfl_xor(q, 1);
      q += __shfl_xor(q, 2);
      q += __shfl_xor(q, 4);
      q += __shfl_xor(q, 8);
      float q_avg = q * (1.0f / 16.0f);

      // ---- fused 15-tap gamma routing (in-register ring buffer) ----
      #pragma unroll
      for (int j = LENF - 1; j > 0; --j) ring[j] = ring[j - 1];
      ring[0] = q_avg;
      float stream = 0.0f;
      #pragma unroll
      for (int j = 0; j < LENF; ++j) stream = fmaf(w[j], ring[j], stream);

      if (m == 0) out[(size_t)t * B_TOT + b] = stream;
    }
  }
}

extern "C" void kernel_launch(void* const* d_in, const int* in_sizes, int n_in,
                              void* d_out, int out_size, void* d_ws, size_t ws_size,
                              hipStream_t stream) {
  const float* x_phy      = (const float*)d_in[0];   // (730, 2048, 2) f32
  const float* phy_static = (const float*)d_in[1];   // (2048, 210)   f32
  float* out = (float*)d_out;                        // (730, 2048, 1) f32
  (void)in_sizes; (void)n_in; (void)out_size; (void)d_ws; (void)ws_size;

  dim3 grid(B_TOT / BPB);       // 128 blocks
  dim3 block(THREADS);          // 8 waves: 16 basins x 16 members per block
  size_t shmem = 2 * BUF_FLOATS * sizeof(float);     // 8 KB double buffer
  hipLaunchKernelGGL(cfe_scan_kernel, grid, block, shmem, stream,
                     x_phy, phy_static, out);
}